// GraphConvolutionTorch_12635793785115
// MI455X (gfx1250) — compile-verified
//
#include <hip/hip_runtime.h>
#include <hip/hip_bf16.h>

typedef __attribute__((ext_vector_type(2))) float v2f;
typedef __attribute__((ext_vector_type(8))) float v8f;

#define WT_STRIDE 68   // padded: bank = (4n + k) & 63 -> conflict-free b64 reads

// ---------------------------------------------------------------------------
// Kernel 1: h = x @ W   (M=N_nodes, K=64, N=64) via V_WMMA_F32_16X16X4_F32
// 256 threads = 8 waves. Each wave computes a 16x64 tile of h.
// W staged TRANSPOSED in LDS so each B fragment is one aligned ds_load_b64.
// ---------------------------------------------------------------------------
__global__ void gcn_gemm(const float* __restrict__ x,
                         const float* __restrict__ W,
                         float* __restrict__ h,
                         int n_rows) {
    __shared__ float sWT[64 * WT_STRIDE];
    for (int i = threadIdx.x; i < 64 * 64; i += 256) {
        int k = i >> 6;          // W row (K index)
        int n = i & 63;          // W col (N index)
        sWT[n * WT_STRIDE + k] = W[i];
    }
    __syncthreads();

    const int wave = threadIdx.x >> 5;       // 0..7
    const int lane = threadIdx.x & 31;       // wave32
    const long mtile = (long)blockIdx.x * 8 + wave;
    if (mtile * 16 >= n_rows) return;        // wave-uniform: EXEC stays all-1s

    const int m     = lane & 15;             // matrix row (A) / col (B) per lane
    const int khalf = (lane >> 4) * 2;       // lanes 0-15 -> K+0,1 ; 16-31 -> K+2,3
    const float* __restrict__ xrow = x + (mtile * 16 + m) * 64 + khalf;

    // Per-lane B base pointers: column (nt*16+m), K offset khalf
    const float* bbase0 = &sWT[(0 * 16 + m) * WT_STRIDE + khalf];
    const float* bbase1 = &sWT[(1 * 16 + m) * WT_STRIDE + khalf];
    const float* bbase2 = &sWT[(2 * 16 + m) * WT_STRIDE + khalf];
    const float* bbase3 = &sWT[(3 * 16 + m) * WT_STRIDE + khalf];

    v8f acc0{}, acc1{}, acc2{}, acc3{};

    #pragma unroll
    for (int kk = 0; kk < 16; ++kk) {        // K advances 4 per WMMA
        const int ko = kk * 4;               // immediate offset (dwords)
        v2f a = *(const v2f*)(xrow + ko);    // global_load_b64, offset kk*16B
        v2f b0 = *(const v2f*)(bbase0 + ko); // ds_load_b64, offset kk*16B
        v2f b1 = *(const v2f*)(bbase1 + ko);
        v2f b2 = *(const v2f*)(bbase2 + ko);
        v2f b3 = *(const v2f*)(bbase3 + ko);
        acc0 = __builtin_amdgcn_wmma_f32_16x16x4_f32(false, a, false, b0,
                                                     (short)0, acc0, false, false);
        acc1 = __builtin_amdgcn_wmma_f32_16x16x4_f32(false, a, false, b1,
                                                     (short)0, acc1, false, false);
        acc2 = __builtin_amdgcn_wmma_f32_16x16x4_f32(false, a, false, b2,
                                                     (short)0, acc2, false, false);
        acc3 = __builtin_amdgcn_wmma_f32_16x16x4_f32(false, a, false, b3,
                                                     (short)0, acc3, false, false);
    }

    // C/D layout: VGPR r -> M=r (lanes 0-15), M=r+8 (lanes 16-31); N = lane&15
    const long rbase = mtile * 16 + (lane >> 4) * 8;
    const int  ncol  = lane & 15;
    #pragma unroll
    for (int r = 0; r < 8; ++r) {
        float* hr = h + (rbase + r) * 64 + ncol;
        hr[0]  = acc0[r];
        hr[16] = acc1[r];
        hr[32] = acc2[r];
        hr[48] = acc3[r];
    }
}

// ---------------------------------------------------------------------------
// Kernel 2: zero the output accumulator, seed deg with 1.0 (self-loop)
// ---------------------------------------------------------------------------
__global__ void gcn_init(float* __restrict__ out, float* __restrict__ deg,
                         int n_nodes, int total) {
    int i = blockIdx.x * blockDim.x + threadIdx.x;
    if (i < total) out[i] = 0.0f;
    if (i < n_nodes) deg[i] = 1.0f;
}

// ---------------------------------------------------------------------------
// Kernel 3: degree on target index (col = edge_index[1])
// ---------------------------------------------------------------------------
__global__ void gcn_deg(const int* __restrict__ ei, float* __restrict__ deg,
                        int n_edges) {
    int e = blockIdx.x * blockDim.x + threadIdx.x;
    if (e < n_edges) {
        atomicAdd(&deg[ei[n_edges + e]], 1.0f);
    }
}

// ---------------------------------------------------------------------------
// Kernel 4: dinv = deg > 0 ? rsqrt(deg) : 0
// ---------------------------------------------------------------------------
__global__ void gcn_dinv(const float* __restrict__ deg, float* __restrict__ dinv,
                         int n_nodes) {
    int i = blockIdx.x * blockDim.x + threadIdx.x;
    if (i < n_nodes) {
        float d = deg[i];
        dinv[i] = (d > 0.0f) ? __frsqrt_rn(d) : 0.0f;
    }
}

// ---------------------------------------------------------------------------
// Kernel 5: edge scatter. One wave32 per edge; lane handles 2 floats.
// Coalesced 256B gather of h[row], float atomics into out[col].
// ---------------------------------------------------------------------------
__global__ void gcn_scatter(const int* __restrict__ ei,
                            const float* __restrict__ h,
                            const float* __restrict__ dinv,
                            float* __restrict__ out,
                            int n_edges) {
    const int wave = threadIdx.x >> 5;
    const int lane = threadIdx.x & 31;
    const long e = (long)blockIdx.x * 8 + wave;
    if (e >= n_edges) return;

    const int r = ei[e];             // source node
    const int c = ei[n_edges + e];   // target node
    const float norm = dinv[r] * dinv[c];

    const float* __restrict__ hs = h + (long)r * 64 + lane * 2;
    float a0 = hs[0];
    float a1 = hs[1];
    float* __restrict__ dst = out + (long)c * 64 + lane * 2;
    atomicAdd(&dst[0], norm * a0);
    atomicAdd(&dst[1], norm * a1);
}

// ---------------------------------------------------------------------------
// Kernel 6: out = relu(out + dinv[i]^2 * h[i,:] + b)   (self-loop + bias + act)
// ---------------------------------------------------------------------------
__global__ void gcn_finalize(float* __restrict__ out,
                             const float* __restrict__ h,
                             const float* __restrict__ dinv,
                             const float* __restrict__ b,
                             int total) {
    int idx = blockIdx.x * blockDim.x + threadIdx.x;
    if (idx < total) {
        int i = idx >> 6;
        int col = idx & 63;
        float s = dinv[i];
        float v = out[idx] + s * s * h[idx] + b[col];
        out[idx] = v > 0.0f ? v : 0.0f;
    }
}

// ---------------------------------------------------------------------------
// Host-side launch
// Inputs (setup_inputs order): x [N,64] f32, edge_index [2,E] i32,
//                              W [64,64] f32, b [64] f32
// Workspace layout: h (N*64 f32) | deg (N f32) | dinv (N f32)
// ---------------------------------------------------------------------------
extern "C" void kernel_launch(void* const* d_in, const int* in_sizes, int n_in,
                              void* d_out, int out_size, void* d_ws, size_t ws_size,
                              hipStream_t stream) {
    const float* x  = (const float*)d_in[0];
    const int*   ei = (const int*)d_in[1];
    const float* W  = (const float*)d_in[2];
    const float* b  = (const float*)d_in[3];

    const int N = in_sizes[0] / 64;   // 100000
    const int E = in_sizes[1] / 2;    // 1000000
    const int total = N * 64;

    float* out  = (float*)d_out;
    float* h    = (float*)d_ws;
    float* deg  = h + (size_t)N * 64;
    float* dinv = deg + N;

    // 1. h = x @ W  (WMMA f32 16x16x4); 8 M-tiles of 16 rows per block
    gcn_gemm<<<(N + 127) / 128, 256, 0, stream>>>(x, W, h, N);

    // 2. zero accumulator, seed degree with self-loop
    gcn_init<<<(total + 255) / 256, 256, 0, stream>>>(out, deg, N, total);

    // 3. degree scatter over edges
    gcn_deg<<<(E + 255) / 256, 256, 0, stream>>>(ei, deg, E);

    // 4. dinv = rsqrt(deg)
    gcn_dinv<<<(N + 255) / 256, 256, 0, stream>>>(deg, dinv, N);

    // 5. normalized feature scatter (wave per edge)
    gcn_scatter<<<(E + 7) / 8, 256, 0, stream>>>(ei, h, dinv, out, E);

    // 6. self-loop term + bias + relu
    gcn_finalize<<<(total + 255) / 256, 256, 0, stream>>>(out, h, dinv, b, total);
}